// CustomSTFT_63333587747523
// MI455X (gfx1250) — compile-verified
//
#include <hip/hip_runtime.h>
#include <math.h>

typedef __attribute__((ext_vector_type(2))) float v2f;
typedef __attribute__((ext_vector_type(8))) float v8f;

#define NBATCH   32
#define TLEN     480000
#define NFFT     800
#define HOP      200
#define FREQ     401
#define FPAD     404                 // sr/si K-dim padded to multiple of 4
#define FPADB    416                 // dft matrices padded to 26 full 16-wide tiles
#define LSTR     412                 // LDS row stride (mod 64 = 28 -> conflict-free lanes 0..15)
#define NFRAMES  2401
#define PADLEN   (TLEN + NFFT)       // 480800 (reflect-padded length)
#define PADOFF   (NFFT / 2)          // 400

#define TTILES   151                 // ceil(2401/16)
#define NPAIRS_F 13                  // pairs of 16-wide freq tiles (26 tiles)
#define NPAIRS_N 25                  // pairs of 16-wide sample tiles (50 tiles)

static __device__ __forceinline__ int reflect_idx(int p) {
  int s = p - PADOFF;
  s = (s < 0) ? -s : s;
  s = (s >= TLEN) ? (2 * TLEN - 2 - s) : s;
  return s;
}

// ---------------------------------------------------------------------------
// Prep 1: pad DFT matrices (800 x 401) -> (800 x 416), zero-filled columns.
// ---------------------------------------------------------------------------
__global__ __launch_bounds__(256)
void pad_dft_kernel(const float* __restrict__ dre, const float* __restrict__ dimg,
                    float* __restrict__ dreP, float* __restrict__ dimP) {
  const int i = blockIdx.x * 256 + threadIdx.x;
  if (i >= NFFT * FPADB) return;
  const int k = i / FPADB;
  const int n = i - k * FPADB;
  float vr = 0.0f, vi = 0.0f;
  if (n < FREQ) { vr = dre[k * FREQ + n]; vi = dimg[k * FREQ + n]; }
  dreP[i] = vr;
  dimP[i] = vi;
}

// ---------------------------------------------------------------------------
// Prep 2: transpose IDFT matrices (800 x 401)[n][f] -> (404 x 800)[f][n],
// zero rows f >= 401.  Makes inverse-GEMM B loads lane-consecutive.
// ---------------------------------------------------------------------------
__global__ __launch_bounds__(256)
void transpose_idft_kernel(const float* __restrict__ ire, const float* __restrict__ iim,
                           float* __restrict__ ireT, float* __restrict__ iimT) {
  const int i = blockIdx.x * 256 + threadIdx.x;
  if (i >= FPAD * NFFT) return;
  const int f = i / NFFT;
  const int n = i - f * NFFT;
  float vr = 0.0f, vi = 0.0f;
  if (f < FREQ) { vr = ire[n * FREQ + f]; vi = iim[n * FREQ + f]; }
  ireT[i] = vr;
  iimT[i] = vi;
}

// ---------------------------------------------------------------------------
// Kernel A: framing (reflect pad fused) + STFT GEMM + mag/phase elementwise.
// Branchless K-loop: A from LDS sliding window, B from padded DFT tables.
// Stores sr and -si into workspace, layout (b, t, f) with f-stride FPAD.
// ---------------------------------------------------------------------------
__global__ __launch_bounds__(256)
void stft_kernel(const float* __restrict__ x,
                 const float* __restrict__ dreP,
                 const float* __restrict__ dimP,
                 float* __restrict__ wsr,
                 float* __restrict__ wsi) {
  const int t0   = blockIdx.x * 16;
  const int b    = blockIdx.y;
  const int tid  = threadIdx.x;
  const int lane = tid & 31;
  const int wave = tid >> 5;

  __shared__ float win[16 * HOP + NFFT];  // 4000 floats: samples for 16 frames

  const float* xb = x + (size_t)b * TLEN;
  for (int i = tid; i < 16 * HOP + NFFT; i += 256) {
    const int p = t0 * HOP + i;
    win[i] = (p < PADLEN) ? xb[reflect_idx(p)] : 0.0f;
  }
  __syncthreads();

  const int m      = lane & 15;      // A row / B column / D column index
  const int half   = lane >> 4;      // 0: K0/K1 | 1: K2/K3
  const int koff   = half * 2;
  const int rowoff = half * 8;       // D rows: M = r + 8*half

  for (int pr = wave; pr < NPAIRS_F; pr += 8) {
    const int n0a = pr * 32;
    const int n0b = n0a + 16;
    const int na  = n0a + m;         // < 416 always: padded table, no guard
    const int nb  = n0b + m;

    v8f acc_ra = {}, acc_ia = {}, acc_rb = {}, acc_ib = {};

    #pragma unroll 2
    for (int k = 0; k < NFFT; k += 4) {
      const int kr = k + koff;
      v2f a;
      a.x = win[m * HOP + kr];
      a.y = win[m * HOP + kr + 1];

      const int o0 = kr * FPADB;
      v2f bra, bia, brb, bib;
      bra.x = dreP[o0 + na];  bra.y = dreP[o0 + FPADB + na];
      bia.x = dimP[o0 + na];  bia.y = dimP[o0 + FPADB + na];
      brb.x = dreP[o0 + nb];  brb.y = dreP[o0 + FPADB + nb];
      bib.x = dimP[o0 + nb];  bib.y = dimP[o0 + FPADB + nb];

      acc_ra = __builtin_amdgcn_wmma_f32_16x16x4_f32(false, a, false, bra, (short)0, acc_ra, false, false);
      acc_ia = __builtin_amdgcn_wmma_f32_16x16x4_f32(false, a, false, bia, (short)0, acc_ia, false, false);
      acc_rb = __builtin_amdgcn_wmma_f32_16x16x4_f32(false, a, false, brb, (short)0, acc_rb, false, false);
      acc_ib = __builtin_amdgcn_wmma_f32_16x16x4_f32(false, a, false, bib, (short)0, acc_ib, false, false);
    }

    #pragma unroll
    for (int tile = 0; tile < 2; ++tile) {
      v8f cr = tile ? acc_rb : acc_ra;
      v8f ci = tile ? acc_ib : acc_ia;
      const int n = (tile ? n0b : n0a) + m;
      #pragma unroll
      for (int r = 0; r < 8; ++r) {
        const int t = t0 + rowoff + r;
        if (t >= NFRAMES) continue;
        const size_t o = (size_t)(b * NFRAMES + t) * FPAD + n;
        if (n < FREQ) {
          float re  = cr[r], im = ci[r];
          float mag = sqrtf(re * re + im * im + 1e-14f);
          float ph  = atan2f(im, re);
          if ((fabsf(im) <= 1e-8f + 1e-4f * mag) && (re < 0.0f))
            ph = 3.14159265358979323846f;
          wsr[o] = mag * cosf(ph);
          wsi[o] = -(mag * sinf(ph));   // negate: inverse GEMM accumulates both
        } else if (n < FPAD) {
          wsr[o] = 0.0f;                // K-pad columns -> exact zeros
          wsi[o] = 0.0f;
        }
      }
    }
  }
}

// ---------------------------------------------------------------------------
// Kernel B: inverse GEMM  rec[b,t,n] = sum_f sr*idftT_re[f,n] + (-si)*idftT_im[f,n]
// One block per (t-tile, batch): sr/si tile staged in LDS once (coalesced),
// each wave processes pairs of 16-wide n-tiles -> 4 WMMAs per shared A fetch.
// ---------------------------------------------------------------------------
__global__ __launch_bounds__(256)
void istft_kernel(const float* __restrict__ wsr,
                  const float* __restrict__ wsi,
                  const float* __restrict__ ireT,
                  const float* __restrict__ iimT,
                  float* __restrict__ rec) {
  const int t0   = blockIdx.x * 16;
  const int b    = blockIdx.y;
  const int tid  = threadIdx.x;
  const int lane = tid & 31;
  const int wave = tid >> 5;

  __shared__ float lsr[16 * LSTR];
  __shared__ float lsi[16 * LSTR];

  for (int i = tid; i < 16 * LSTR; i += 256) {
    const int row = i / LSTR;
    const int col = i - row * LSTR;
    const int t   = t0 + row;
    float vr = 0.0f, vi = 0.0f;
    if (t < NFRAMES && col < FPAD) {
      const size_t o = (size_t)(b * NFRAMES + t) * FPAD + col;
      vr = wsr[o];
      vi = wsi[o];
    }
    lsr[i] = vr;
    lsi[i] = vi;
  }
  __syncthreads();

  const int m      = lane & 15;
  const int half   = lane >> 4;
  const int koff   = half * 2;
  const int rowoff = half * 8;

  for (int p = wave; p < NPAIRS_N; p += 8) {
    const int n0a = p * 32;
    const int n0b = n0a + 16;
    const int na  = n0a + m;          // < 800 always
    const int nb  = n0b + m;

    v8f acca = {}, accb = {};

    #pragma unroll 2
    for (int k = 0; k < FPAD; k += 4) {
      const int f = k + koff;         // max 403 < FPAD rows of ireT
      v2f ar, ai;
      ar.x = lsr[m * LSTR + f];  ar.y = lsr[m * LSTR + f + 1];
      ai.x = lsi[m * LSTR + f];  ai.y = lsi[m * LSTR + f + 1];

      const int o0 = f * NFFT;
      v2f bra, bia, brb, bib;
      bra.x = ireT[o0 + na];  bra.y = ireT[o0 + NFFT + na];
      bia.x = iimT[o0 + na];  bia.y = iimT[o0 + NFFT + na];
      brb.x = ireT[o0 + nb];  brb.y = ireT[o0 + NFFT + nb];
      bib.x = iimT[o0 + nb];  bib.y = iimT[o0 + NFFT + nb];

      acca = __builtin_amdgcn_wmma_f32_16x16x4_f32(false, ar, false, bra, (short)0, acca, false, false);
      acca = __builtin_amdgcn_wmma_f32_16x16x4_f32(false, ai, false, bia, (short)0, acca, false, false);
      accb = __builtin_amdgcn_wmma_f32_16x16x4_f32(false, ar, false, brb, (short)0, accb, false, false);
      accb = __builtin_amdgcn_wmma_f32_16x16x4_f32(false, ai, false, bib, (short)0, accb, false, false);
    }

    #pragma unroll
    for (int tile = 0; tile < 2; ++tile) {
      v8f acc = tile ? accb : acca;
      const int n = (tile ? n0b : n0a) + m;
      #pragma unroll
      for (int r = 0; r < 8; ++r) {
        const int t = t0 + rowoff + r;
        if (t < NFRAMES)
          rec[(size_t)(b * NFRAMES + t) * NFFT + n] = acc[r];
      }
    }
  }
}

// ---------------------------------------------------------------------------
// Kernel C: deterministic overlap-add gather (<=4 frames per sample) + crop.
// ---------------------------------------------------------------------------
__global__ __launch_bounds__(256)
void ola_kernel(const float* __restrict__ rec, float* __restrict__ out) {
  const int i = blockIdx.x * 256 + threadIdx.x;
  if (i >= NBATCH * TLEN) return;
  const int b = i / TLEN;
  const int p = i - b * TLEN;
  const int j = p + PADOFF;                        // padded coordinate
  int t_hi = j / HOP;                              if (t_hi > NFRAMES - 1) t_hi = NFRAMES - 1;
  int t_lo = (j - (NFFT - 1) + (HOP - 1)) / HOP;   if (t_lo < 0) t_lo = 0;

  const float* rb = rec + (size_t)b * NFRAMES * NFFT;
  float acc = 0.0f;
  for (int t = t_lo; t <= t_hi; ++t)
    acc += rb[(size_t)t * NFFT + (j - t * HOP)];
  out[i] = acc;
}

// ---------------------------------------------------------------------------
extern "C" void kernel_launch(void* const* d_in, const int* in_sizes, int n_in,
                              void* d_out, int out_size, void* d_ws, size_t ws_size,
                              hipStream_t stream) {
  const float* x    = (const float*)d_in[0];   // (32, 480000)
  const float* dre  = (const float*)d_in[1];   // (800, 401)
  const float* dimg = (const float*)d_in[2];   // (800, 401)
  const float* ire  = (const float*)d_in[3];   // (800, 401)
  const float* iim  = (const float*)d_in[4];   // (800, 401)
  float* out = (float*)d_out;                  // (32, 1, 480000)

  // Workspace layout (floats)
  float* w = (float*)d_ws;
  const size_t SZ_DFT  = (size_t)NFFT * FPADB;       //   332,800 each
  const size_t SZ_IDFT = (size_t)FPAD * NFFT;        //   323,200 each
  const size_t SZ_SPEC = (size_t)NBATCH * NFRAMES * FPAD;   // 31,040,128 each
  float* dreP = w;                   w += SZ_DFT;
  float* dimP = w;                   w += SZ_DFT;
  float* ireT = w;                   w += SZ_IDFT;
  float* iimT = w;                   w += SZ_IDFT;
  float* wsr  = w;                   w += SZ_SPEC;
  float* wsi  = w;                   w += SZ_SPEC;
  float* rec  = w;                   // 32*2401*800 = 61,465,600

  // Prep: padded / transposed basis tables (tiny; L2-resident afterwards)
  {
    const int n1 = NFFT * FPADB;
    pad_dft_kernel<<<(n1 + 255) / 256, 256, 0, stream>>>(dre, dimg, dreP, dimP);
    const int n2 = FPAD * NFFT;
    transpose_idft_kernel<<<(n2 + 255) / 256, 256, 0, stream>>>(ire, iim, ireT, iimT);
  }

  dim3 gA(TTILES, NBATCH, 1);
  stft_kernel<<<gA, 256, 0, stream>>>(x, dreP, dimP, wsr, wsi);

  dim3 gB(TTILES, NBATCH, 1);
  istft_kernel<<<gB, 256, 0, stream>>>(wsr, wsi, ireT, iimT, rec);

  const int total = NBATCH * TLEN;
  ola_kernel<<<(total + 255) / 256, 256, 0, stream>>>(rec, out);
}